// WordPooling_49151605736122
// MI455X (gfx1250) — compile-verified
//
#include <hip/hip_runtime.h>
#include <hip/hip_bf16.h>

typedef __attribute__((ext_vector_type(2))) float v2f;
typedef __attribute__((ext_vector_type(8))) float v8f;

// ---------------------------------------------------------------------------
// WMMA path, H fixed at compile time (H=1024 for the reference shape).
//
// One wave32 computes one 16x16 output tile via 16 chained
// V_WMMA_F32_16X16X4_F32 ops:  D += A_j (16x4, row j = 0.25) * B_j (4x16).
// Result: D[m][n] = 0.25 * sum_{k<4} x[64*tw + 4m + k][16*th + n]  (exact f32).
//
// Addressing: one 32-bit per-lane element offset from the uniform base
// pointer -> backend selects GVS mode (saddr + voffset + 24-bit imm), so all
// 32 loads / 8 stores use immediate offsets with no per-access VALU.
// ---------------------------------------------------------------------------
template <int HC>
__global__ __launch_bounds__(256) void WordPool_wmma_fixedH(
    const float* __restrict__ x,   // [W*4, HC]
    float* __restrict__ out,       // [W,   HC]
    int tilesW, int tilesH)
{
    const int wave   = threadIdx.x >> 5;                 // 8 waves / block
    const int lane   = threadIdx.x & 31;
    const long tile  = (long)blockIdx.x * 8 + wave;
    const long total = (long)tilesW * tilesH;
    if (tile >= total) return;                           // wave-uniform: EXEC all-1s

    const unsigned tw = (unsigned)(tile / tilesH);       // word-tile index
    const unsigned th = (unsigned)(tile % tilesH);       // hidden-tile index

    const unsigned m    = lane & 15;                     // M (and N) index
    const unsigned half = (unsigned)lane >> 4;           // K-half select
    const float scale = 0.25f;                           // 1/span, exact

    // 32-bit element offset of row (64*tw + 2*half), col (16*th + m).
    const unsigned base = (tw * 64u + 2u * half) * (unsigned)HC + th * 16u + m;

    v8f c = {};
#pragma unroll
    for (int j = 0; j < 16; ++j) {
        const float aw = (m == (unsigned)j) ? scale : 0.0f;  // A_j: row j only
        v2f a; a.x = aw; a.y = aw;

        // B_j: V0 = rows {4j, 4j+2}, V1 = rows {4j+1, 4j+3} (lane-half split).
        v2f b;
        b.x = __builtin_nontemporal_load(x + (base + (unsigned)(j * 4 * HC)));
        b.y = __builtin_nontemporal_load(x + (base + (unsigned)(j * 4 * HC + HC)));

        c = __builtin_amdgcn_wmma_f32_16x16x4_f32(
                false, a, false, b, (short)0, c, false, false);
    }

    // D: VGPR r -> M = r + 8*half, N = lane&15. 32-bit offset + imm stores.
    const unsigned obase = (tw * 16u + half * 8u) * (unsigned)HC + th * 16u + m;
#pragma unroll
    for (int r = 0; r < 8; ++r)
        __builtin_nontemporal_store(c[r], out + (obase + (unsigned)(r * HC)));
}

// ---------------------------------------------------------------------------
// Generic scalar fallback: arbitrary contiguous span / shapes (still a
// single-pass, coalesced, bandwidth-bound kernel).
// ---------------------------------------------------------------------------
__global__ __launch_bounds__(256) void WordPool_generic(
    const float* __restrict__ x, float* __restrict__ out,
    int H, long totalOut, int span)
{
    long i = (long)blockIdx.x * blockDim.x + threadIdx.x;
    if (i >= totalOut) return;
    const long w = i / H;
    const long h = i - w * H;
    float s = 0.0f;
    for (int k = 0; k < span; ++k)
        s += __builtin_nontemporal_load(x + (w * span + k) * (long)H + h);
    out[i] = s / (float)span;
}

// ---------------------------------------------------------------------------
// Host launcher. Geometry derived from flat sizes:
//   H = |hidden|/|seg|, W = out_size/H, span = |seg|/W.
// Dispatch is a pure function of the sizes -> deterministic across calls.
// ---------------------------------------------------------------------------
extern "C" void kernel_launch(void* const* d_in, const int* in_sizes, int n_in,
                              void* d_out, int out_size, void* d_ws, size_t ws_size,
                              hipStream_t stream) {
    const float* x = (const float*)d_in[0];
    float* out     = (float*)d_out;

    const long nHS  = in_sizes[0];            // B*S*H
    const long T    = in_sizes[1];            // B*S tokens
    const int  H    = (int)(nHS / T);
    const int  W    = (int)((long)out_size / H);
    const int  span = (int)(T / (long)W);

    if (span == 4 && H == 1024 && (W % 16) == 0) {
        const int tilesW = W / 16;
        const int tilesH = H / 16;                       // 64
        const long tiles = (long)tilesW * tilesH;        // 32768 for ref shape
        const int blocks = (int)((tiles + 7) / 8);       // 8 waves (tiles) / block
        WordPool_wmma_fixedH<1024><<<blocks, 256, 0, stream>>>(x, out, tilesW, tilesH);
    } else {
        const long totalOut = (long)W * H;
        const int blocks = (int)((totalOut + 255) / 256);
        WordPool_generic<<<blocks, 256, 0, stream>>>(x, out, H, totalOut, span);
    }
}